// SparseAttention_77464030151052
// MI455X (gfx1250) — compile-verified
//
#include <hip/hip_runtime.h>
#include <hip/hip_bf16.h>

typedef float v2f __attribute__((ext_vector_type(2)));
typedef float v8f __attribute__((ext_vector_type(8)));

#define BATCH 2
#define HEADS 8
#define SEQ_L 2048
#define SEQ_S 2048
#define EDIM  64
#define DDIM  64
#define TILE_M 16
#define NWAVES 8
#define SC_STRIDE 2052   // 2048 + 4 floats pad: 8-row offset lands on bank +32
#define O_STRIDE  68     // 64 + 4 floats pad

__device__ inline v8f wmma_f32_4(v2f a, v2f b, v8f c) {
  // V_WMMA_F32_16X16X4_F32 : D(16x16 f32) = A(16x4 f32) * B(4x16 f32) + C
  return __builtin_amdgcn_wmma_f32_16x16x4_f32(
      false, a, false, b, (short)0, c, false, false);
}

__device__ inline unsigned sortkey(float s) {
  unsigned u = __float_as_uint(s);
  return (u & 0x80000000u) ? ~u : (u | 0x80000000u);
}

__global__ __launch_bounds__(256, 2)
void SparseAttention_77464030151052_kernel(const float* __restrict__ q,
                                           const float* __restrict__ kk,
                                           const float* __restrict__ vv,
                                           float* __restrict__ out,
                                           int ktop) {
  __shared__ float    sc[TILE_M * SC_STRIDE];     // scores, then probabilities
  __shared__ unsigned hist[TILE_M][256];          // radix-select histograms
  __shared__ float    o_lds[TILE_M * O_STRIDE];   // output accumulation tile
  __shared__ float    rs_den[TILE_M];             // softmax denominators

  const int tile = blockIdx.x;      // 0..127 query tile
  const int h    = blockIdx.y;      // 0..7
  const int b    = blockIdx.z;      // 0..1
  const int lane = threadIdx.x & 31;
  const int w    = threadIdx.x >> 5;

  const float SCALE = 0.125f;       // 1/sqrt(64)

  // zero output accumulation tile
  for (int i = threadIdx.x; i < TILE_M * O_STRIDE; i += 256) o_lds[i] = 0.0f;

  // ---------------- Phase 1: scores = Q_tile(16xE) * K^T -> LDS ----------------
  const int m     = lane & 15;          // M / N index within 16
  const int ehalf = (lane >> 4) * 2;    // K-dim half select per ISA A/B layout

  v2f qa[16];
#pragma unroll
  for (int ks = 0; ks < 16; ++ks) {
    int e = ks * 4 + ehalf;
    const float* qp = &q[(((size_t)b * SEQ_L + (size_t)tile * 16 + m) * HEADS + h) * EDIM + e];
    qa[ks].x = qp[0];
    qa[ks].y = qp[1];
  }

  for (int ch = w; ch < SEQ_S / 16; ch += NWAVES) {
    int n = ch * 16 + m;              // key index for this lane's B columns
    v8f acc = {};
#pragma unroll
    for (int ks = 0; ks < 16; ++ks) {
      int e = ks * 4 + ehalf;
      const float* kp = &kk[(((size_t)b * SEQ_S + n) * HEADS + h) * EDIM + e];
      v2f bf;
      bf.x = kp[0];
      bf.y = kp[1];
      acc = wmma_f32_4(qa[ks], bf, acc);
    }
    // C layout: vgpr i -> row (i + 8*(lane>>4)), col (lane&15)
    int rbase = 8 * (lane >> 4);
    int col   = ch * 16 + m;
#pragma unroll
    for (int i = 0; i < 8; ++i)
      sc[(rbase + i) * SC_STRIDE + col] = acc[i];
  }
  __syncthreads();

  // ------------- Phase 2+3: per-row radix select + softmax weights -------------
  // wave w owns rows 2w and 2w+1
  for (int rr = 0; rr < 2; ++rr) {
    const int r = w * 2 + rr;
    unsigned prefix = 0, prefmask = 0;
    int krem = ktop;
    float rmax = -3.402823e38f;

#pragma unroll
    for (int lvl = 0; lvl < 4; ++lvl) {
      const int sh = 24 - 8 * lvl;
      for (int i = lane; i < 256; i += 32) hist[r][i] = 0u;
      __threadfence_block();
      for (int it = 0; it < SEQ_S / 32; ++it) {
        int j = it * 32 + lane;
        float s = sc[r * SC_STRIDE + j];
        unsigned u = sortkey(s);
        if (lvl == 0) rmax = fmaxf(rmax, s);
        if ((u & prefmask) == prefix)
          atomicAdd(&hist[r][(u >> sh) & 255u], 1u);
      }
      __threadfence_block();
      if (lvl == 0) {
        for (int off = 16; off; off >>= 1)
          rmax = fmaxf(rmax, __shfl_xor(rmax, off, 32));
      }
      // all lanes redundantly scan bins high->low (LDS broadcast reads)
      unsigned cum = 0;
      int sel = 0;
      for (int bin = 255; bin >= 0; --bin) {
        unsigned c = hist[r][bin];
        if ((int)(cum + c) >= krem) { sel = bin; break; }
        cum += c;
      }
      krem -= (int)cum;
      prefix  |= ((unsigned)sel) << sh;
      prefmask |= 0xFFu << sh;
    }
    const unsigned thresh = prefix;   // exact bit pattern of k-th largest

    // convert kept scores to unnormalized softmax weights, in place
    float den = 0.0f;
    for (int it = 0; it < SEQ_S / 32; ++it) {
      int j = it * 32 + lane;
      float s = sc[r * SC_STRIDE + j];
      float p = 0.0f;
      if (sortkey(s) >= thresh) p = __expf(SCALE * (s - rmax));
      sc[r * SC_STRIDE + j] = p;
      den += p;
    }
    for (int off = 16; off; off >>= 1) den += __shfl_xor(den, off, 32);
    if (lane == 0) rs_den[r] = den;
  }
  __syncthreads();

  // ---------------- Phase 4: O_tile(16xD) = P(16xS) * V(SxD) ----------------
  {
    const int s0 = w * (SEQ_S / NWAVES);     // 256-key slice per wave
    v8f zero = {};
    v8f accs[4];
#pragma unroll
    for (int t = 0; t < 4; ++t) accs[t] = zero;

    for (int ks = 0; ks < SEQ_S / NWAVES; ks += 4) {
      int c0 = s0 + ks + ehalf;
      v2f af;
      af.x = sc[m * SC_STRIDE + c0];
      af.y = sc[m * SC_STRIDE + c0 + 1];
#pragma unroll
      for (int t = 0; t < 4; ++t) {
        int n  = t * 16 + m;
        const float* vp = &vv[(((size_t)b * SEQ_S + c0) * HEADS + h) * DDIM + n];
        v2f bf;
        bf.x = vp[0];
        bf.y = vp[HEADS * DDIM];    // next key row
        accs[t] = wmma_f32_4(af, bf, accs[t]);
      }
    }
    int rbase = 8 * (lane >> 4);
#pragma unroll
    for (int t = 0; t < 4; ++t)
#pragma unroll
      for (int i = 0; i < 8; ++i)
        atomicAdd(&o_lds[(rbase + i) * O_STRIDE + t * 16 + m], accs[t][i]);
  }
  __syncthreads();

  // ---------------- Phase 5: normalize + store ----------------
  for (int i = threadIdx.x; i < TILE_M * DDIM; i += 256) {
    int mr = i >> 6;
    int d  = i & 63;
    float val = o_lds[mr * O_STRIDE + d] / rs_den[mr];
    int row = tile * 16 + mr;
    out[(((size_t)b * SEQ_L + row) * HEADS + h) * DDIM + d] = val;
  }
}

extern "C" void kernel_launch(void* const* d_in, const int* in_sizes, int n_in,
                              void* d_out, int out_size, void* d_ws, size_t ws_size,
                              hipStream_t stream) {
  (void)in_sizes; (void)n_in; (void)d_ws; (void)ws_size; (void)out_size;
  const float* q  = (const float*)d_in[0];
  const float* kk = (const float*)d_in[1];
  const float* vv = (const float*)d_in[2];
  float* out = (float*)d_out;

  int ktop = (int)(SEQ_S * 0.6);   // 1228
  if (ktop < 2) ktop = 2;

  dim3 grid(SEQ_L / TILE_M, HEADS, BATCH);
  dim3 block(256, 1, 1);
  SparseAttention_77464030151052_kernel<<<grid, block, 0, stream>>>(q, kk, vv, out, ktop);
}